// PatchQuantumGenerator_53901839564947
// MI455X (gfx1250) — compile-verified
//
#include <hip/hip_runtime.h>
#include <hip/hip_bf16.h>
#include <math.h>

#define BATCH     16384
#define NQ        8
#define NGEN      4
#define PATCH     64
#define QDEPTH    6
#define PI_F      3.14159265358979323846f

typedef __attribute__((ext_vector_type(2))) float v2f;
typedef __attribute__((ext_vector_type(8))) float v8f;

__device__ __forceinline__ float lrelu(float x) { return x >= 0.f ? x : 0.2f * x; }

// ---------------------------------------------------------------------------
// Kernel 1: conditioning MLP.  One wave32 per batch row; activations in LDS.
// ---------------------------------------------------------------------------
__global__ void pre_mlp(const float* __restrict__ labels,
                        const float* __restrict__ W1, const float* __restrict__ b1,
                        const float* __restrict__ W2, const float* __restrict__ b2,
                        const float* __restrict__ Wa, const float* __restrict__ ba,
                        const float* __restrict__ Wp1, const float* __restrict__ bp1,
                        const float* __restrict__ Wp2, const float* __restrict__ bp2,
                        float* __restrict__ angles, float* __restrict__ cpost)
{
    __shared__ float sh [8][64];
    __shared__ float sh2[8][64];
    __shared__ float sp [8][32];

    const int wv   = threadIdx.x >> 5;
    const int lane = threadIdx.x & 31;
    const int b    = blockIdx.x * 8 + wv;
    const float* lab = labels + b * 10;

    // h = lrelu(labels @ W1^T + b1)
    #pragma unroll
    for (int t = 0; t < 2; ++t) {
        int o = lane + 32 * t;
        float s = b1[o];
        #pragma unroll
        for (int k = 0; k < 10; ++k) s += lab[k] * W1[o * 10 + k];
        sh[wv][o] = lrelu(s);
    }
    __syncthreads();

    // h2 = h + lrelu(h @ W2^T + b2)
    #pragma unroll
    for (int t = 0; t < 2; ++t) {
        int o = lane + 32 * t;
        float s = b2[o];
        for (int k = 0; k < 64; ++k) s += sh[wv][k] * W2[o * 64 + k];
        sh2[wv][o] = sh[wv][o] + lrelu(s);
    }
    __syncthreads();

    // angles = tanh(h2 @ Wa^T + ba) * pi
    if (lane < 16) {
        float s = ba[lane];
        for (int k = 0; k < 64; ++k) s += sh2[wv][k] * Wa[lane * 64 + k];
        angles[b * 16 + lane] = tanhf(s) * PI_F;
    }

    // cpost = lrelu(lrelu(labels @ Wp1^T + bp1) @ Wp2^T + bp2)
    {
        float s = bp1[lane];
        #pragma unroll
        for (int k = 0; k < 10; ++k) s += lab[k] * Wp1[lane * 10 + k];
        sp[wv][lane] = lrelu(s);
    }
    __syncthreads();
    {
        float s = bp2[lane];
        for (int k = 0; k < 32; ++k) s += sp[wv][k] * Wp2[lane * 32 + k];
        cpost[b * 32 + lane] = lrelu(s);
    }
}

// ---------------------------------------------------------------------------
// Kernel 2: 8-qubit statevector sim.  One wave32 per (batch, generator).
// Flat amp index = lane*8 + r ; wire i <-> bit (7-i).
//   wires 0..4 -> lane bits 4..0  (RY via __shfl_xor)
//   wires 5..7 -> register bits 2..0 (pure VALU)
// RZ / CZ are diagonal -> local only.
// ---------------------------------------------------------------------------
template <int W>
__device__ __forceinline__ void apply_ry(float (&re)[8], float (&im)[8],
                                         float th, int lane)
{
    float s = __sinf(0.5f * th);
    float c = __cosf(0.5f * th);
    if constexpr (W <= 4) {
        constexpr int lb = 4 - W;
        float sg = ((lane >> lb) & 1) ? s : -s;
        #pragma unroll
        for (int r = 0; r < 8; ++r) {
            float pre = __shfl_xor(re[r], 1 << lb, 32);
            float pim = __shfl_xor(im[r], 1 << lb, 32);
            re[r] = __builtin_fmaf(c, re[r], sg * pre);
            im[r] = __builtin_fmaf(c, im[r], sg * pim);
        }
    } else {
        constexpr int m = 1 << (7 - W);
        #pragma unroll
        for (int r0 = 0; r0 < 8; ++r0) {
            if ((r0 & m) == 0) {
                const int r1 = r0 | m;
                float ar = re[r0], ai = im[r0];
                float br = re[r1], bi = im[r1];
                re[r0] = c * ar - s * br;  im[r0] = c * ai - s * bi;
                re[r1] = s * ar + c * br;  im[r1] = s * ai + c * bi;
            }
        }
    }
}

template <int W>
__device__ __forceinline__ void apply_rz(float (&re)[8], float (&im)[8],
                                         float th, int lane)
{
    float s = __sinf(0.5f * th);
    float c = __cosf(0.5f * th);
    if constexpr (W <= 4) {
        float p = ((lane >> (4 - W)) & 1) ? s : -s;  // phase = c + i*p
        #pragma unroll
        for (int r = 0; r < 8; ++r) {
            float nr = c * re[r] - p * im[r];
            im[r]    = c * im[r] + p * re[r];
            re[r]    = nr;
        }
    } else {
        constexpr int m = 1 << (7 - W);
        #pragma unroll
        for (int r = 0; r < 8; ++r) {
            float p = (r & m) ? s : -s;
            float nr = c * re[r] - p * im[r];
            im[r]    = c * im[r] + p * re[r];
            re[r]    = nr;
        }
    }
}

template <int W1>  // CZ on wires (W1, W1+1)
__device__ __forceinline__ void apply_cz(float (&re)[8], float (&im)[8], int lane)
{
    if constexpr (W1 + 1 <= 4) {                  // both lane bits
        int cond = ((lane >> (4 - W1)) & 1) & ((lane >> (3 - W1)) & 1);
        float sg = cond ? -1.f : 1.f;
        #pragma unroll
        for (int r = 0; r < 8; ++r) { re[r] *= sg; im[r] *= sg; }
    } else if constexpr (W1 <= 4) {               // W1 lane bit, W1+1 register bit
        float sg = ((lane >> (4 - W1)) & 1) ? -1.f : 1.f;
        constexpr int m = 1 << (7 - (W1 + 1));
        #pragma unroll
        for (int r = 0; r < 8; ++r)
            if (r & m) { re[r] *= sg; im[r] *= sg; }
    } else {                                      // both register bits
        constexpr int m1 = 1 << (7 - W1), m2 = 1 << (7 - (W1 + 1));
        #pragma unroll
        for (int r = 0; r < 8; ++r)
            if ((r & m1) && (r & m2)) { re[r] = -re[r]; im[r] = -im[r]; }
    }
}

__global__ void qcirc(const float* __restrict__ noise,
                      const float* __restrict__ angles,
                      const float* __restrict__ qp,
                      float* __restrict__ qf)
{
    const int lane = threadIdx.x & 31;
    const int wid  = (blockIdx.x << 3) | (threadIdx.x >> 5);
    const int b    = wid >> 2;
    const int g    = wid & 3;

    const float* nz = noise  + b * NQ;
    const float* an = angles + b * 16;
    const float* wp = qp     + g * (QDEPTH * NQ);

    float re[8], im[8];
    #pragma unroll
    for (int r = 0; r < 8; ++r) { re[r] = 0.f; im[r] = 0.f; }
    re[0] = (lane == 0) ? 1.f : 0.f;

    // noise encoding RY
    apply_ry<0>(re, im, nz[0], lane); apply_ry<1>(re, im, nz[1], lane);
    apply_ry<2>(re, im, nz[2], lane); apply_ry<3>(re, im, nz[3], lane);
    apply_ry<4>(re, im, nz[4], lane); apply_ry<5>(re, im, nz[5], lane);
    apply_ry<6>(re, im, nz[6], lane); apply_ry<7>(re, im, nz[7], lane);
    // class-conditioning RY
    apply_ry<0>(re, im, an[0], lane); apply_ry<1>(re, im, an[1], lane);
    apply_ry<2>(re, im, an[2], lane); apply_ry<3>(re, im, an[3], lane);
    apply_ry<4>(re, im, an[4], lane); apply_ry<5>(re, im, an[5], lane);
    apply_ry<6>(re, im, an[6], lane); apply_ry<7>(re, im, an[7], lane);
    // class-conditioning RZ
    apply_rz<0>(re, im, an[8],  lane); apply_rz<1>(re, im, an[9],  lane);
    apply_rz<2>(re, im, an[10], lane); apply_rz<3>(re, im, an[11], lane);
    apply_rz<4>(re, im, an[12], lane); apply_rz<5>(re, im, an[13], lane);
    apply_rz<6>(re, im, an[14], lane); apply_rz<7>(re, im, an[15], lane);

    for (int d = 0; d < QDEPTH; ++d) {
        const float* wd = wp + d * NQ;
        apply_ry<0>(re, im, wd[0], lane); apply_ry<1>(re, im, wd[1], lane);
        apply_ry<2>(re, im, wd[2], lane); apply_ry<3>(re, im, wd[3], lane);
        apply_ry<4>(re, im, wd[4], lane); apply_ry<5>(re, im, wd[5], lane);
        apply_ry<6>(re, im, wd[6], lane); apply_ry<7>(re, im, wd[7], lane);
        apply_cz<0>(re, im, lane); apply_cz<1>(re, im, lane);
        apply_cz<2>(re, im, lane); apply_cz<3>(re, im, lane);
        apply_cz<4>(re, im, lane); apply_cz<5>(re, im, lane);
        apply_cz<6>(re, im, lane);
    }

    // p0/max(p0): the global prob-sum cancels -> probs[:64] / max(probs[:64])
    float pr[8];
    float pmax = 0.f;
    #pragma unroll
    for (int r = 0; r < 8; ++r) {
        pr[r] = re[r] * re[r] + im[r] * im[r];
        pmax  = fmaxf(pmax, pr[r]);
    }
    if (lane >= 8) pmax = 0.f;           // first 64 amps live in lanes 0..7
    #pragma unroll
    for (int off = 16; off > 0; off >>= 1)
        pmax = fmaxf(pmax, __shfl_xor(pmax, off, 32));

    if (lane < 8) {
        float inv = 1.f / pmax;
        float* dst = qf + b * (NGEN * PATCH) + g * PATCH + lane * 8;
        #pragma unroll
        for (int r = 0; r < 8; ++r) dst[r] = pr[r] * inv;
    }
}

// ---------------------------------------------------------------------------
// Kernel 3: post-MLP with V_WMMA_F32_16X16X4_F32.
// One workgroup (8 waves) owns 16 batch rows; each wave computes one 16x16
// output tile; LayerNorm done in LDS between GEMMs.
// ---------------------------------------------------------------------------
#define XS 289   // 288 + pad (odd multiple keeps 64-bank conflicts away)
#define YS 129   // 128 + pad

__global__ void post_mlp(const float* __restrict__ qf, const float* __restrict__ cpost,
                         const float* __restrict__ Wq1, const float* __restrict__ bq1,
                         const float* __restrict__ g1,  const float* __restrict__ be1,
                         const float* __restrict__ Wq2, const float* __restrict__ bq2,
                         const float* __restrict__ g2,  const float* __restrict__ be2,
                         const float* __restrict__ Wq3, const float* __restrict__ bq3,
                         float* __restrict__ out)
{
    __shared__ float xs[16 * XS];
    __shared__ float ys[16 * YS];
    __shared__ float zs[16 * YS];
    __shared__ float mu_s[16], rs_s[16];

    const int tid  = threadIdx.x;
    const int wv   = tid >> 5;
    const int lane = tid & 31;
    const int mrow = lane & 15;     // M (A) / N (B) sub-index
    const int kg   = lane >> 4;     // K group: lanes 0-15 -> K+0/1, 16-31 -> K+2/3
    const int b0   = blockIdx.x * 16;

    // stage x = concat(qf, cpost) into LDS (16 x 288)
    for (int idx = tid; idx < 16 * 288; idx += 256) {
        int r = idx / 288, c = idx - r * 288;
        float v = (c < 256) ? qf[(b0 + r) * 256 + c] : cpost[(b0 + r) * 32 + (c - 256)];
        xs[r * XS + c] = v;
    }
    __syncthreads();

    // ---- GEMM1: (16x288) @ Wq1^T(288x128) -> lrelu -> ys ----
    {
        v8f acc = {0.f, 0.f, 0.f, 0.f, 0.f, 0.f, 0.f, 0.f};
        const int   ncol = 16 * wv + mrow;
        const float* wr  = Wq1 + ncol * 288;
        for (int k0 = 0; k0 < 288; k0 += 4) {
            int ka = k0 + 2 * kg;
            v2f a;  a.x  = xs[mrow * XS + ka];  a.y  = xs[mrow * XS + ka + 1];
            v2f bb; bb.x = wr[ka];              bb.y = wr[ka + 1];
            acc = __builtin_amdgcn_wmma_f32_16x16x4_f32(false, a, false, bb,
                                                        (short)0, acc, false, false);
        }
        float bias = bq1[ncol];
        #pragma unroll
        for (int j = 0; j < 8; ++j) {
            int row = j + 8 * kg;
            ys[row * YS + ncol] = lrelu(acc[j] + bias);
        }
    }
    __syncthreads();

    // ---- LN1 ----
    if (tid < 16) {
        float s = 0.f, ss = 0.f;
        for (int c = 0; c < 128; ++c) { float v = ys[tid * YS + c]; s += v; ss += v * v; }
        float mu = s * (1.f / 128.f);
        float var = ss * (1.f / 128.f) - mu * mu;
        mu_s[tid] = mu;  rs_s[tid] = rsqrtf(var + 1e-5f);
    }
    __syncthreads();
    for (int idx = tid; idx < 16 * 128; idx += 256) {
        int r = idx >> 7, c = idx & 127;
        ys[r * YS + c] = (ys[r * YS + c] - mu_s[r]) * rs_s[r] * g1[c] + be1[c];
    }
    __syncthreads();

    // ---- GEMM2: (16x128) @ Wq2^T(128x128) -> lrelu -> zs ----
    {
        v8f acc = {0.f, 0.f, 0.f, 0.f, 0.f, 0.f, 0.f, 0.f};
        const int   ncol = 16 * wv + mrow;
        const float* wr  = Wq2 + ncol * 128;
        for (int k0 = 0; k0 < 128; k0 += 4) {
            int ka = k0 + 2 * kg;
            v2f a;  a.x  = ys[mrow * YS + ka];  a.y  = ys[mrow * YS + ka + 1];
            v2f bb; bb.x = wr[ka];              bb.y = wr[ka + 1];
            acc = __builtin_amdgcn_wmma_f32_16x16x4_f32(false, a, false, bb,
                                                        (short)0, acc, false, false);
        }
        float bias = bq2[ncol];
        #pragma unroll
        for (int j = 0; j < 8; ++j) {
            int row = j + 8 * kg;
            zs[row * YS + ncol] = lrelu(acc[j] + bias);
        }
    }
    __syncthreads();

    // ---- LN2 ----
    if (tid < 16) {
        float s = 0.f, ss = 0.f;
        for (int c = 0; c < 128; ++c) { float v = zs[tid * YS + c]; s += v; ss += v * v; }
        float mu = s * (1.f / 128.f);
        float var = ss * (1.f / 128.f) - mu * mu;
        mu_s[tid] = mu;  rs_s[tid] = rsqrtf(var + 1e-5f);
    }
    __syncthreads();
    for (int idx = tid; idx < 16 * 128; idx += 256) {
        int r = idx >> 7, c = idx & 127;
        zs[r * YS + c] = (zs[r * YS + c] - mu_s[r]) * rs_s[r] * g2[c] + be2[c];
    }
    __syncthreads();

    // ---- GEMM3: (16x128) @ Wq3^T(128x64) -> tanh -> out (waves 0..3) ----
    if (wv < 4) {
        v8f acc = {0.f, 0.f, 0.f, 0.f, 0.f, 0.f, 0.f, 0.f};
        const int   ncol = 16 * wv + mrow;
        const float* wr  = Wq3 + ncol * 128;
        for (int k0 = 0; k0 < 128; k0 += 4) {
            int ka = k0 + 2 * kg;
            v2f a;  a.x  = zs[mrow * YS + ka];  a.y  = zs[mrow * YS + ka + 1];
            v2f bb; bb.x = wr[ka];              bb.y = wr[ka + 1];
            acc = __builtin_amdgcn_wmma_f32_16x16x4_f32(false, a, false, bb,
                                                        (short)0, acc, false, false);
        }
        float bias = bq3[ncol];
        #pragma unroll
        for (int j = 0; j < 8; ++j) {
            int row = j + 8 * kg;
            out[(b0 + row) * 64 + ncol] = tanhf(acc[j] + bias);
        }
    }
}

// ---------------------------------------------------------------------------
extern "C" void kernel_launch(void* const* d_in, const int* in_sizes, int n_in,
                              void* d_out, int out_size, void* d_ws, size_t ws_size,
                              hipStream_t stream)
{
    const float* noise  = (const float*)d_in[0];
    const float* labels = (const float*)d_in[1];
    const float* qprm   = (const float*)d_in[2];
    const float* W1  = (const float*)d_in[3];  const float* b1  = (const float*)d_in[4];
    const float* W2  = (const float*)d_in[5];  const float* b2  = (const float*)d_in[6];
    const float* Wa  = (const float*)d_in[7];  const float* ba  = (const float*)d_in[8];
    const float* Wp1 = (const float*)d_in[9];  const float* bp1 = (const float*)d_in[10];
    const float* Wp2 = (const float*)d_in[11]; const float* bp2 = (const float*)d_in[12];
    const float* Wq1 = (const float*)d_in[13]; const float* bq1 = (const float*)d_in[14];
    const float* g1  = (const float*)d_in[15]; const float* be1 = (const float*)d_in[16];
    const float* Wq2 = (const float*)d_in[17]; const float* bq2 = (const float*)d_in[18];
    const float* g2  = (const float*)d_in[19]; const float* be2 = (const float*)d_in[20];
    const float* Wq3 = (const float*)d_in[21]; const float* bq3 = (const float*)d_in[22];

    float* angles = (float*)d_ws;                  // BATCH * 16
    float* cpostw = angles + (size_t)BATCH * 16;   // BATCH * 32
    float* qfw    = cpostw + (size_t)BATCH * 32;   // BATCH * 256

    pre_mlp<<<BATCH / 8, 256, 0, stream>>>(labels, W1, b1, W2, b2, Wa, ba,
                                           Wp1, bp1, Wp2, bp2, angles, cpostw);
    qcirc<<<(BATCH * NGEN) / 8, 256, 0, stream>>>(noise, angles, qprm, qfw);
    post_mlp<<<BATCH / 16, 256, 0, stream>>>(qfw, cpostw,
                                             Wq1, bq1, g1, be1,
                                             Wq2, bq2, g2, be2,
                                             Wq3, bq3, (float*)d_out);
}